// Model_541165879869
// MI455X (gfx1250) — compile-verified
//
#include <hip/hip_runtime.h>
#include <hip/hip_bf16.h>
#include <math.h>

// Problem constants
constexpr int B_ = 4, T_ = 336, N_ = 862, D_ = 64, R_ = 8, K_ = 6, P_ = 96;
constexpr int NP  = 864;   // N padded to multiple of 16 (WMMA M/K)
constexpr int TP  = 384;   // T padded to multiple of 64 (4-wide t-strips in GEMM1)
constexpr int CAP = 32;    // max stored sparse entries per adjacency row
constexpr int KC  = 16;    // K-chunk staged to LDS in GEMM2

typedef __attribute__((ext_vector_type(2))) float v2f;
typedef __attribute__((ext_vector_type(8))) float v8f;

// CDNA5 async global->LDS copy, 16B per lane. VDST = per-lane LDS byte offset
// (low 32 bits of the generic LDS address per ISA flat-aperture rules).
__device__ __forceinline__ void async_copy_b128(const float* gsrc, float* lds_generic) {
    unsigned lds = (unsigned)(size_t)lds_generic;
    unsigned long long ga = (unsigned long long)(size_t)gsrc;
    asm volatile("global_load_async_to_lds_b128 %0, %1, off" :: "v"(lds), "v"(ga) : "memory");
}
__device__ __forceinline__ void wait_asynccnt0() {
    asm volatile("s_wait_asynccnt 0x0" ::: "memory");
}

// ---------------------------------------------------------------------------
// Stage 0: time-mean of x_enc:  xbar[b,n] = (1/T) sum_t x_enc[b,t,n]
__global__ void k_xbar(const float* __restrict__ x, float* __restrict__ xbar) {
    int id = blockIdx.x * blockDim.x + threadIdx.x;
    if (id >= B_ * N_) return;
    int b = id / N_, n = id % N_;
    const float* p = x + (size_t)b * T_ * N_ + n;
    float s = 0.f;
    for (int t = 0; t < T_; ++t) s += p[(size_t)t * N_];
    xbar[id] = s * (1.0f / T_);
}

// ---------------------------------------------------------------------------
// Stage 1: tiny precomputes.  scal = {alpha,beta,gamma,delta,s,g,1/sqrt(R),1+s}
__global__ void k_scalars(const float* __restrict__ W_embed, const float* __restrict__ b_embed,
                          const float* __restrict__ W1, const float* __restrict__ W2,
                          const float* __restrict__ gate, const float* __restrict__ W_gcn,
                          const float* __restrict__ g_gcn,
                          float* __restrict__ w, float* __restrict__ bvec,
                          float* __restrict__ wg, float* __restrict__ bg,
                          float* __restrict__ scal) {
    __shared__ float su1[R_], su2[R_], sc1[R_], sc2[R_];
    int t = threadIdx.x;
    if (t < D_) { w[t] = W_embed[t]; bvec[t] = b_embed[t]; }
    if (t < R_) {
        float u1 = 0, u2 = 0, c1 = 0, c2 = 0;
        for (int d = 0; d < D_; ++d) {
            u1 += W_embed[d] * W1[d * R_ + t];
            u2 += W_embed[d] * W2[d * R_ + t];
            c1 += b_embed[d] * W1[d * R_ + t];
            c2 += b_embed[d] * W2[d * R_ + t];
        }
        su1[t] = u1; su2[t] = u2; sc1[t] = c1; sc2[t] = c2;
    }
    if (t < D_) {
        float a = 0, c = 0;
        for (int d = 0; d < D_; ++d) {
            float v = W_gcn[d * D_ + t];
            a += W_embed[d] * v;
            c += b_embed[d] * v;
        }
        wg[t] = a; bg[t] = c;
    }
    __syncthreads();
    if (t == 0) {
        float al = 0, be = 0, ga = 0, de = 0;
        for (int r = 0; r < R_; ++r) {
            al += su1[r] * su2[r]; be += su1[r] * sc2[r];
            ga += sc1[r] * su2[r]; de += sc1[r] * sc2[r];
        }
        float s = 1.0f / (1.0f + expf(-gate[0]));
        scal[0] = al; scal[1] = be; scal[2] = ga; scal[3] = de;
        scal[4] = s;  scal[5] = g_gcn[0];
        scal[6] = rsqrtf((float)R_); scal[7] = 1.0f + s;
    }
}

__global__ void k_dinv(const float* __restrict__ A_prior, float* __restrict__ dinv) {
    int n = blockIdx.x * blockDim.x + threadIdx.x;
    if (n >= N_) return;
    const float* row = A_prior + (size_t)n * N_;
    float s = 0.f;
    for (int m = 0; m < N_; ++m) s += row[m];
    dinv[n] = (s > 0.f) ? rsqrtf(s) : 0.f;
}

__global__ void k_ahat(const float* __restrict__ A_prior, const float* __restrict__ dinv,
                       float* __restrict__ Ahat) {
    int id = blockIdx.x * blockDim.x + threadIdx.x;
    if (id >= NP * NP) return;
    int n = id / NP, m = id % NP;
    float v = 0.f;
    if (n < N_ && m < N_) v = A_prior[(size_t)n * N_ + m] * dinv[n] * dinv[m];
    Ahat[id] = v;
}

__global__ void k_rn(const float* __restrict__ Ahat, float* __restrict__ rn) {
    int n = blockIdx.x * blockDim.x + threadIdx.x;
    if (n >= N_) return;
    const float* row = Ahat + (size_t)n * NP;
    float s = 0.f;
    for (int m = 0; m < NP; ++m) s += row[m];
    rn[n] = s;
}

__global__ void k_wqp(const float* __restrict__ W_head, const float* __restrict__ w,
                      const float* __restrict__ wg, const float* __restrict__ scal,
                      float* __restrict__ Wq, float* __restrict__ Wp) {
    int id = blockIdx.x * blockDim.x + threadIdx.x;
    if (id >= T_ * P_) return;
    int t = id / P_, p = id % P_;
    const float* Wh = W_head + ((size_t)t * D_) * P_ + p;
    float a = 0, c = 0;
    for (int d = 0; d < D_; ++d) {
        float v = Wh[(size_t)d * P_];
        a += w[d] * v;
        c += wg[d] * v;
    }
    Wq[id] = a;
    Wp[id] = scal[5] * c;   // fold g_gcn in
}

__global__ void k_cb(const float* __restrict__ W_head, const float* __restrict__ bvec,
                     const float* __restrict__ bg, float* __restrict__ Cb, float* __restrict__ Cbg) {
    int p = blockIdx.x * blockDim.x + threadIdx.x;
    if (p >= P_) return;
    float a = 0, c = 0;
    for (int t = 0; t < T_; ++t)
        for (int d = 0; d < D_; ++d) {
            float v = W_head[((size_t)t * D_ + d) * P_ + p];
            a += bvec[d] * v;
            c += bg[d] * v;
        }
    Cb[p] = a; Cbg[p] = c;
}

// ---------------------------------------------------------------------------
// Stage 2: per adjacency row — exact top-K(6) threshold (tie-counting) on
// closed-form logits, renormalized softmax weights, ballot-ordered compaction.
__global__ void k_topk(const float* __restrict__ xbar, const float* __restrict__ scal,
                       int* __restrict__ cnt, int* __restrict__ idxs, float* __restrict__ wgts) {
    int row = blockIdx.x;            // b*NP + n
    int b = row / NP, n = row % NP;
    int lane = threadIdx.x;
    if (n >= N_) { if (lane == 0) cnt[row] = 0; return; }
    const float alpha = scal[0], beta = scal[1], gamma = scal[2], delta = scal[3], isr = scal[6];
    const float xn = xbar[b * N_ + n];
    const int CH = (NP + 31) / 32;   // 27
    float lg[CH];
#pragma unroll
    for (int i = 0; i < CH; ++i) {
        int m = i * 32 + lane;
        float xm = (m < N_) ? xbar[b * N_ + m] : 0.0f;
        float l = (alpha * xn * xm + beta * xn + gamma * xm + delta) * isr;
        lg[i] = (m < N_) ? l : -__builtin_inff();
    }
    float prev = __builtin_inff(), thresh = -__builtin_inff(), rowmax = -__builtin_inff();
    int needed = K_;
    for (int it = 0; it < K_ && needed > 0; ++it) {
        float lm = -__builtin_inff();
#pragma unroll
        for (int i = 0; i < CH; ++i) if (lg[i] < prev) lm = fmaxf(lm, lg[i]);
        for (int off = 16; off; off >>= 1) lm = fmaxf(lm, __shfl_xor(lm, off));
        if (it == 0) rowmax = lm;
        int c = 0;
#pragma unroll
        for (int i = 0; i < CH; ++i) c += (lg[i] == lm) ? 1 : 0;
        for (int off = 16; off; off >>= 1) c += __shfl_xor(c, off);
        needed -= c; prev = lm; thresh = lm;
    }
    float se = 0.f;
#pragma unroll
    for (int i = 0; i < CH; ++i) if (lg[i] >= thresh) se += expf(lg[i] - rowmax);
    for (int off = 16; off; off >>= 1) se += __shfl_xor(se, off);
    const float inv = 1.0f / se;
    int base = 0;
    int*   myi = idxs + (size_t)row * CAP;
    float* myw = wgts + (size_t)row * CAP;
#pragma unroll
    for (int i = 0; i < CH; ++i) {
        int m = i * 32 + lane;
        bool sel = (m < N_) && (lg[i] >= thresh);
        unsigned bits = (unsigned)__ballot(sel);
        if (sel) {
            int pos = base + __popc(bits & ((1u << lane) - 1u));
            if (pos < CAP) { myi[pos] = m; myw[pos] = expf(lg[i] - rowmax) * inv; }
        }
        base += __popc(bits);
    }
    if (lane == 0) cnt[row] = (base < CAP) ? base : CAP;
}

// ---------------------------------------------------------------------------
// Stage 3: q[b,n,t] = x[b,t,n] + s * sum_j w_j x[b,t,idx_j]; zero-pad to TP.
__global__ void k_qrow(const float* __restrict__ x, const int* __restrict__ cnt,
                       const int* __restrict__ idxs, const float* __restrict__ wgts,
                       const float* __restrict__ scal, float* __restrict__ Q) {
    int row = blockIdx.x; int b = row / NP, n = row % NP;
    float* q = Q + (size_t)row * TP;
    if (n >= N_) { for (int t = threadIdx.x; t < TP; t += blockDim.x) q[t] = 0.f; return; }
    const float s = scal[4];
    int c = cnt[row];
    __shared__ int   si[CAP];
    __shared__ float sw[CAP];
    if (threadIdx.x < CAP) {
        si[threadIdx.x] = (threadIdx.x < c) ? idxs[(size_t)row * CAP + threadIdx.x] : 0;
        sw[threadIdx.x] = (threadIdx.x < c) ? wgts[(size_t)row * CAP + threadIdx.x] : 0.f;
    }
    __syncthreads();
    const float* xb = x + (size_t)b * T_ * N_;
    for (int t = threadIdx.x; t < TP; t += blockDim.x) {
        float val = 0.f;
        if (t < T_) {
            const float* xt = xb + (size_t)t * N_;
            float y = 0.f;
            for (int j = 0; j < c; ++j) y += sw[j] * xt[si[j]];
            val = xt[n] + s * y;
        }
        q[t] = val;
    }
}

// ---------------------------------------------------------------------------
// Stage 4 (WMMA fp32): P[b] (NP x TP) = Ahat (NP x NP) @ Q[b] (NP x TP).
// One wave per 16x64 output strip: 4 accumulators reuse each A-fragment 4x.
__global__ void k_gemm1(const float* __restrict__ Ahat, const float* __restrict__ Q,
                        float* __restrict__ Pm) {
    const int NT = NP / 16, TG = TP / 64;     // 54, 6
    int tid = blockIdx.x;
    int b = tid / (NT * TG); int r = tid % (NT * TG);
    int n0 = (r / TG) * 16, tg = (r % TG) * 64;
    int lane = threadIdx.x, half = lane >> 4, lm = lane & 15;
    v8f acc[4] = {};
    const float* Arow = Ahat + (size_t)(n0 + lm) * NP;
    const float* Qb   = Q + (size_t)b * NP * TP;
    for (int k0 = 0; k0 < NP; k0 += 32) {
        __builtin_prefetch(Arow + k0 + 32, 0, 1);   // global_prefetch_b8 next A panel
#pragma unroll
        for (int kk = 0; kk < 32; kk += 4) {
            int k = k0 + kk;
            v2f a = *(const v2f*)(Arow + k + 2 * half);          // A[lm][k+2h+j]
            const float* q0 = Qb + (size_t)(k + 2 * half) * TP + tg + lm;
#pragma unroll
            for (int j = 0; j < 4; ++j) {
                v2f bf;
                bf.x = q0[j * 16];                               // B[k+2h+0][col]
                bf.y = q0[TP + j * 16];                          // B[k+2h+1][col]
                acc[j] = __builtin_amdgcn_wmma_f32_16x16x4_f32(false, a, false, bf,
                                                               (short)0, acc[j], false, false);
            }
        }
    }
    float* Pb = Pm + (size_t)b * NP * TP;
#pragma unroll
    for (int j = 0; j < 4; ++j)
#pragma unroll
        for (int v = 0; v < 8; ++v)
            Pb[(size_t)(n0 + 8 * half + v) * TP + tg + j * 16 + lm] = acc[j][v];
}

// ---------------------------------------------------------------------------
// Stage 5 (WMMA fp32): Out(B*NP x P) = Q @ Wq + P @ (g*Wp) + epilogue,
// stored transposed as (B, P, N). 4 waves/block share B-panels staged in LDS
// via CDNA5 async global->LDS loads (ASYNCcnt + barrier).
__global__ void k_gemm2(const float* __restrict__ Q, const float* __restrict__ Pm,
                        const float* __restrict__ Wq, const float* __restrict__ Wp,
                        const float* __restrict__ Cb, const float* __restrict__ Cbg,
                        const float* __restrict__ rn, const float* __restrict__ b_head,
                        const float* __restrict__ scal, float* __restrict__ out) {
    __shared__ float sB[KC * P_];                 // 16 x 96 panel chunk (6 KB)
    int wave = threadIdx.x >> 5;
    int lane = threadIdx.x & 31, half = lane >> 4, lm = lane & 15;
    int m0 = (blockIdx.x * 4 + wave) * 16;        // this wave's 16 output rows
    v8f acc[6] = {};
#pragma unroll
    for (int pass = 0; pass < 2; ++pass) {
        const float* panel = pass ? Wp : Wq;
        const float* arow  = (pass ? Pm : Q) + (size_t)(m0 + lm) * TP;
        for (int k0 = 0; k0 < T_; k0 += KC) {
            // cooperative async stage of panel[k0..k0+16) x 96 into LDS
            const float* src = panel + (size_t)k0 * P_;
#pragma unroll
            for (int it = 0; it < (KC * P_) / (128 * 4); ++it) {   // 3 iters
                int e = (it * 128 + threadIdx.x) * 4;
                async_copy_b128(src + e, &sB[e]);
            }
            wait_asynccnt0();
            __syncthreads();
#pragma unroll
            for (int kk = 0; kk < KC; kk += 4) {
                v2f a = *(const v2f*)(arow + k0 + kk + 2 * half);
#pragma unroll
                for (int j = 0; j < 6; ++j) {
                    v2f bf;
                    bf.x = sB[(kk + 2 * half + 0) * P_ + j * 16 + lm];
                    bf.y = sB[(kk + 2 * half + 1) * P_ + j * 16 + lm];
                    acc[j] = __builtin_amdgcn_wmma_f32_16x16x4_f32(false, a, false, bf,
                                                                   (short)0, acc[j], false, false);
                }
            }
            __syncthreads();   // before next chunk overwrites sB
        }
    }
    const float s1 = scal[7], g = scal[5];
#pragma unroll
    for (int j = 0; j < 6; ++j) {
        int p = j * 16 + lm;
        const float cb = Cb[p], cbg = Cbg[p], bh = b_head[p];
#pragma unroll
        for (int v = 0; v < 8; ++v) {
            int mrow = m0 + 8 * half + v;
            int b = mrow / NP, n = mrow % NP;
            if (n < N_) {
                float val = acc[j][v] + bh + s1 * (cb + g * rn[n] * cbg);
                out[((size_t)b * P_ + p) * N_ + n] = val;
            }
        }
    }
}

// ---------------------------------------------------------------------------
extern "C" void kernel_launch(void* const* d_in, const int* in_sizes, int n_in,
                              void* d_out, int out_size, void* d_ws, size_t ws_size,
                              hipStream_t stream) {
    const float* x_enc   = (const float*)d_in[0];
    const float* W_embed = (const float*)d_in[1];
    const float* b_embed = (const float*)d_in[2];
    const float* W1      = (const float*)d_in[3];
    const float* W2      = (const float*)d_in[4];
    const float* gate    = (const float*)d_in[5];
    const float* A_prior = (const float*)d_in[6];
    const float* W_gcn   = (const float*)d_in[7];
    const float* g_gcn   = (const float*)d_in[8];
    const float* W_head  = (const float*)d_in[9];
    const float* b_head  = (const float*)d_in[10];
    float* out = (float*)d_out;

    float* ws = (float*)d_ws;
    size_t o = 0;
    auto take = [&](size_t n) { float* p = ws + o; o += (n + 3) & ~(size_t)3; return p; };
    float* xbar = take((size_t)B_ * N_);
    float* w    = take(D_);
    float* bvec = take(D_);
    float* wg   = take(D_);
    float* bg   = take(D_);
    float* scal = take(16);
    float* dinv = take(N_);
    float* rn   = take(N_);
    float* Ahat = take((size_t)NP * NP);
    float* Wq   = take((size_t)T_ * P_);
    float* Wp   = take((size_t)T_ * P_);
    float* Cb   = take(P_);
    float* Cbg  = take(P_);
    int*   cnt  = (int*)take((size_t)B_ * NP);
    int*   idxs = (int*)take((size_t)B_ * NP * CAP);
    float* wgts = take((size_t)B_ * NP * CAP);
    float* Q    = take((size_t)B_ * NP * TP);
    float* Pm   = take((size_t)B_ * NP * TP);
    (void)ws_size; (void)in_sizes; (void)n_in; (void)out_size;

    k_scalars<<<1, 64, 0, stream>>>(W_embed, b_embed, W1, W2, gate, W_gcn, g_gcn,
                                    w, bvec, wg, bg, scal);
    k_xbar<<<(B_ * N_ + 127) / 128, 128, 0, stream>>>(x_enc, xbar);
    k_dinv<<<(N_ + 127) / 128, 128, 0, stream>>>(A_prior, dinv);
    k_ahat<<<(NP * NP + 255) / 256, 256, 0, stream>>>(A_prior, dinv, Ahat);
    k_rn<<<(N_ + 127) / 128, 128, 0, stream>>>(Ahat, rn);
    k_wqp<<<(T_ * P_ + 127) / 128, 128, 0, stream>>>(W_head, w, wg, scal, Wq, Wp);
    k_cb<<<(P_ + 63) / 64, 64, 0, stream>>>(W_head, bvec, bg, Cb, Cbg);
    k_topk<<<B_ * NP, 32, 0, stream>>>(xbar, scal, cnt, idxs, wgts);
    k_qrow<<<B_ * NP, 128, 0, stream>>>(x_enc, cnt, idxs, wgts, scal, Q);
    k_gemm1<<<B_ * (NP / 16) * (TP / 64), 32, 0, stream>>>(Ahat, Q, Pm);
    k_gemm2<<<(B_ * NP / 16) / 4, 128, 0, stream>>>(Q, Pm, Wq, Wp, Cb, Cbg, rn,
                                                    b_head, scal, out);
}